// DeepSeekMOE_13950053777726
// MI455X (gfx1250) — compile-verified
//
#include <hip/hip_runtime.h>
#include <hip/hip_bf16.h>

// Grouped masked GEMM for DeepSeekMOE up/gate + down projections.
// fp32 WMMA (v_wmma_f32_16x16x4_f32), one wave32 per 32x16 output tile.
// Memory-bound: weights (~553 MB) streamed once, non-temporal; x stays in L2.
// K is a template constant so the inner loop unrolls with no divergent bound.

typedef __attribute__((ext_vector_type(2))) float v2f;
typedef __attribute__((ext_vector_type(8))) float v8f;

#define G_GROUPS 8
#define M_ROWS   32
#define K_UG     4096
#define N_UG     2816
#define K_DN     1408
#define N_DN     4096
#define N_OUT    (N_UG + N_DN)                 // 6912
#define NT_UG    (N_UG / 16)                   // 176 n-tiles per group (22 blocks x 8 waves)
#define NT_DN    (N_DN / 16)                   // 256 n-tiles per group (32 blocks x 8 waves)

// One wave computes a 32x16 output tile of  out[g] = x[g] @ w[g]^T  (NT layout).
//   grid.y = group g (scalar), grid.x * 8 waves = n-tiles.
// K: reduction depth (compile-time). N_W: rows of w per group. COLBASE: output column offset.
template <int K, int N_W, int COLBASE>
__global__ __launch_bounds__(256) void moe_masked_gemm(
    const float* __restrict__ x, const float* __restrict__ w,
    const int* __restrict__ masked_m, float* __restrict__ out)
{
    const int g    = (int)blockIdx.y;                       // scalar
    const int nt   = (int)blockIdx.x * 8 + (int)(threadIdx.x >> 5);
    const int lane = (int)(threadIdx.x & 31);
    const int half = lane >> 4;    // 0: lanes 0-15, 1: lanes 16-31
    const int l16  = lane & 15;

    const float* xg = x + (size_t)g * M_ROWS * K;
    const float* wg = w + ((size_t)g * N_W + (size_t)nt * 16) * K;

    // Per-lane base pointers matching the 16x4 f32 A layout and 4x16 B layout:
    //   A (16x4): lane<16 holds row=l16, k+{0,1}; lane>=16 holds row=l16, k+{2,3}
    //   B (4x16): VGPR v: K=v (lanes 0-15), K=v+2 (lanes 16-31); col = l16
    const float* aptr0 = xg + (size_t)l16 * K + 2 * half;   // rows 0-15
    const float* aptr1 = aptr0 + (size_t)16 * K;            // rows 16-31
    const float* bptr  = wg + (size_t)l16 * K + 2 * half;   // 16 cols of W (NT layout)

    v8f c0 = {};   // rows 0-15  x cols n0..n0+15
    v8f c1 = {};   // rows 16-31 x cols n0..n0+15

    #pragma unroll 8
    for (int k = 0; k < K; k += 4) {
        v2f a0 = *(const v2f*)(aptr0 + k);                           // hot in L2 (5.4 MB total)
        v2f a1 = *(const v2f*)(aptr1 + k);
        v2f b  = __builtin_nontemporal_load((const v2f*)(bptr + k)); // streaming weights
        // D = A * B + C ; 8-arg form (neg_a, A, neg_b, B, c_mod, C, reuse_a, reuse_b)
        c0 = __builtin_amdgcn_wmma_f32_16x16x4_f32(false, a0, false, b, (short)0, c0, false, false);
        c1 = __builtin_amdgcn_wmma_f32_16x16x4_f32(false, a1, false, b, (short)0, c1, false, false);
    }

    // C/D layout: VGPR v -> row = v + 8*half, col = COLBASE + nt*16 + l16.
    // Apply row-validity mask branch-free (EXEC stays all-1s, required around WMMA).
    const int mm = masked_m[g];
    float* og = out + (size_t)g * M_ROWS * N_OUT + (size_t)COLBASE + (size_t)nt * 16 + l16;
    #pragma unroll
    for (int v = 0; v < 8; ++v) {
        const int r0 = v + 8 * half;
        const int r1 = r0 + 16;
        og[(size_t)r0 * N_OUT] = (r0 < mm) ? c0[v] : 0.0f;
        og[(size_t)r1 * N_OUT] = (r1 < mm) ? c1[v] : 0.0f;
    }
}

extern "C" void kernel_launch(void* const* d_in, const int* in_sizes, int n_in,
                              void* d_out, int out_size, void* d_ws, size_t ws_size,
                              hipStream_t stream) {
    const float* x_ug     = (const float*)d_in[0];
    const float* w_ug     = (const float*)d_in[1];
    const float* x_dn     = (const float*)d_in[2];
    const float* w_dn     = (const float*)d_in[3];
    const int*   masked_m = (const int*)d_in[4];
    float* out = (float*)d_out;

    // up/gate: 176 n-tiles x 8 groups, 8 waves (256 thr) per block -> 22 x 8 blocks
    moe_masked_gemm<K_UG, N_UG, 0>
        <<<dim3(NT_UG / 8, G_GROUPS), dim3(256), 0, stream>>>(x_ug, w_ug, masked_m, out);
    // down: 256 n-tiles x 8 groups -> 32 x 8 blocks, output columns offset by N_UG
    moe_masked_gemm<K_DN, N_DN, N_UG>
        <<<dim3(NT_DN / 8, G_GROUPS), dim3(256), 0, stream>>>(x_dn, w_dn, masked_m, out);
}